// SDF_39118562132124
// MI455X (gfx1250) — compile-verified
//
#include <hip/hip_runtime.h>
#include <hip/hip_bf16.h>
#include <cstdint>
#include <cstddef>

namespace {
constexpr int kNV     = 2048;            // edges / vertices per shape
constexpr int kTPB    = 256;             // 8 wave32s per block
constexpr int kSlices = 8;               // edge-range split within a block
constexpr int kPPB    = kTPB / kSlices;  // 32 points per block (one per lane)
constexpr int kEPS    = kNV / kSlices;   // 256 edges per slice
constexpr int kEPT    = kNV / kTPB;      // 8 edges precomputed per thread
}

// Byte offset of an LDS object relative to LDS_BASE (AS3 pointers are 32-bit).
__device__ __forceinline__ unsigned lds_byte_off(const void* p) {
  return (unsigned)(unsigned long long)(const __attribute__((address_space(3))) void*)p;
}

__global__ __launch_bounds__(kTPB) void SDF_39118562132124_kernel(
    const float* __restrict__ x,      // [B, NPTS, 2]
    const int*   __restrict__ s,      // [B]
    const float* __restrict__ V_all,  // [S, NV, 2]
    const int*   __restrict__ F_all,  // [S, NV, 2]
    float*       __restrict__ out,    // [B, NPTS]
    int npts, int total) {
  // 64 KB LDS. Lifetime-phased:
  //  (1) floats [0, 2*kNV)      : raw V staging (async copy target)
  //  (2) floats [0, 8*kNV)      : per-edge records [v0x,v0y,ex,ey,enx,eny,v1y,pad]
  //  (3) floats [0, kTPB)       : partial dmin;  ints [kTPB, 2*kTPB): partial parity
  __shared__ float lds[kNV * 8];

  const int tid   = threadIdx.x;
  const int slice = tid >> 5;           // tid / kPPB : which edge slice (0..7)
  const int lpt   = tid & (kPPB - 1);   // point within block (0..31)
  const int pt    = blockIdx.x * kPPB + lpt;
  const int ptc   = pt < total ? pt : (total - 1);

  // Batch index is uniform per block (kPPB divides npts).
  const int blk0 = blockIdx.x * kPPB;
  const int b    = (blk0 < total ? blk0 : (total - 1)) / npts;
  const int sb   = s[b];
  const float* Vsel = V_all + (size_t)sb * kNV * 2;
  const int*   Fsel = F_all + (size_t)sb * kNV * 2;

  // ---- Phase 1: async-stage raw V (16 KB) into LDS (CDNA5 async data mover) ----
  {
    const unsigned lbase = lds_byte_off(lds);
    const char* g = (const char*)Vsel;
    #pragma unroll
    for (int i = 0; i < (kNV * 2 * 4) / (kTPB * 16); ++i) {  // 4 x 16 B per thread
      const int c = tid + i * kTPB;
      unsigned    loff = lbase + (unsigned)(c * 16);
      const char* ga   = g + (size_t)c * 16;
      asm volatile("global_load_async_to_lds_b128 %0, %1, off"
                   :: "v"(loff), "v"(ga) : "memory");
    }
    asm volatile("s_wait_asynccnt 0x0" ::: "memory");
  }
  __syncthreads();

  // ---- Phase 2: per-edge precompute (raw V -> regs, barrier, write records) ----
  float v0x[kEPT], v0y[kEPT], v1x[kEPT], v1y[kEPT];
  #pragma unroll
  for (int k = 0; k < kEPT; ++k) {
    const int j  = tid * kEPT + k;
    const int f0 = Fsel[2 * j];
    const int f1 = Fsel[2 * j + 1];
    v0x[k] = lds[2 * f0];  v0y[k] = lds[2 * f0 + 1];
    v1x[k] = lds[2 * f1];  v1y[k] = lds[2 * f1 + 1];
  }
  __syncthreads();   // all raw reads complete before the in-place overwrite
  #pragma unroll
  for (int k = 0; k < kEPT; ++k) {
    const int j = tid * kEPT + k;
    const float ex = v1x[k] - v0x[k];
    const float ey = v1y[k] - v0y[k];
    const float ee = ex * ex + ey * ey;
    float* rec = &lds[8 * j];
    rec[0] = v0x[k];  rec[1] = v0y[k];
    rec[2] = ex;      rec[3] = ey;
    rec[4] = ex / ee; rec[5] = ey / ee;   // en = e / |e|^2 (matches reference)
    rec[6] = v1y[k];  rec[7] = 0.0f;
  }
  __syncthreads();

  // ---- Phase 3: each wave32 sweeps one 256-edge slice for 32 points ----
  const float px = x[(size_t)ptc * 2];
  const float py = x[(size_t)ptc * 2 + 1];
  float dmin = __builtin_inff();
  int   acc  = 0;
  const int j0 = slice * kEPS;
  #pragma unroll 4
  for (int jj = 0; jj < kEPS; ++jj) {
    const float* rec = &lds[8 * (j0 + jj)];   // wave-uniform address -> broadcast
    const float v0yj = rec[1];
    const float exj  = rec[2], eyj = rec[3];
    const float rx = px - rec[0];
    const float ry = py - v0yj;
    float t = rx * rec[4] + ry * rec[5];
    t = fminf(fmaxf(t, 0.0f), 1.0f);
    const float qx = rx - exj * t;
    const float qy = ry - eyj * t;
    const float d  = qx * qx + qy * qy;
    dmin = fminf(dmin, d);
    const bool c0 = py >= v0yj;
    const bool c1 = py <  rec[6];                  // V1.y
    const bool c2 = (eyj * qx - exj * qy) < 0.0f;  // dot(n, p) < 0, n = (ey, -ex)
    acc += (c0 & c1 & c2) ? 1 : 0;
    acc -= ((!c0) & (!c1) & (!c2)) ? 1 : 0;
  }

  // ---- Phase 4: exact cross-slice combine (min of mins, sum of int counts) ----
  __syncthreads();                       // all phase-3 record reads done
  lds[tid] = dmin;                       // pd[slice][lpt] at float index tid
  ((int*)lds)[kTPB + tid] = acc;         // pa[slice][lpt]
  __syncthreads();
  if (tid < kPPB && pt < total) {
    float dm = __builtin_inff();
    int   a  = 0;
    #pragma unroll
    for (int k = 0; k < kSlices; ++k) {
      dm = fminf(dm, lds[k * kPPB + tid]);
      a += ((int*)lds)[kTPB + k * kPPB + tid];
    }
    const float sign = 1.0f - 2.0f * (float)(a & 1);  // == reference's (..)%2 sign
    out[pt] = sign * sqrtf(dm);
  }
}

extern "C" void kernel_launch(void* const* d_in, const int* in_sizes, int n_in,
                              void* d_out, int out_size, void* d_ws, size_t ws_size,
                              hipStream_t stream) {
  (void)n_in; (void)d_ws; (void)ws_size;
  const float* x     = (const float*)d_in[0];  // [B, NPTS, 2]
  const int*   s     = (const int*)  d_in[1];  // [B]
  const float* V_all = (const float*)d_in[2];  // [S, NV, 2]
  const int*   F_all = (const int*)  d_in[3];  // [S, NV, 2]
  float*       out   = (float*)d_out;          // [B, NPTS]

  const int B      = in_sizes[1];
  const int npts   = in_sizes[0] / (B * 2);
  const int total  = out_size;                 // B * NPTS
  const int blocks = (total + kPPB - 1) / kPPB;

  SDF_39118562132124_kernel<<<dim3(blocks), dim3(kTPB), 0, stream>>>(
      x, s, V_all, F_all, out, npts, total);
}